// RecurrentBlock_2877628088935
// MI455X (gfx1250) — compile-verified
//
#include <hip/hip_runtime.h>
#include <math.h>

// Problem dims (fixed by the reference)
#define BSZ 16
#define ESZ 1024
#define SSZ 256

typedef float v4f __attribute__((ext_vector_type(4)));

#define AS_GLOBAL __attribute__((address_space(1)))
#define AS_LDS    __attribute__((address_space(3)))

// ---- CDNA5 async global -> LDS copy (per-lane b32), ASYNCcnt tracked ----
__device__ __forceinline__ void async_copy_b32(const float* g, float* l) {
#if __has_builtin(__builtin_amdgcn_global_load_async_to_lds_b32)
  // Signature (from clang diagnostic): (global int*, local int*, imm offset, imm cpol)
  __builtin_amdgcn_global_load_async_to_lds_b32(
      (AS_GLOBAL int*)g, (AS_LDS int*)l, /*imm offset*/0, /*cpol*/0);
#else
#warning "global_load_async_to_lds_b32 builtin unavailable: synchronous fallback in use"
  *l = *g;
#endif
}

__device__ __forceinline__ void wait_async0() {
#if __has_builtin(__builtin_amdgcn_s_wait_asynccnt)
  __builtin_amdgcn_s_wait_asynccnt(0);
#else
  asm volatile("s_wait_asynccnt 0" ::: "memory");
#endif
}

// One block = one e-row of lambda shared by 4 batches; 256 threads = 8 wave32.
// threadIdx: [b_sub(2b) | s_group(6b)]; each thread owns 4 consecutive s.
__global__ __launch_bounds__(256)
void ssm_update_kernel(const float* __restrict__ xr_, const float* __restrict__ xi_,
                       const float* __restrict__ dr_, const float* __restrict__ di_,
                       const float* __restrict__ Br_, const float* __restrict__ Bi_,
                       const float* __restrict__ hr_, const float* __restrict__ hi_,
                       const float* __restrict__ lr_, const float* __restrict__ li_,
                       float* __restrict__ out_re, float* __restrict__ out_im) {
  __shared__ alignas(16) float s_lr[SSZ];
  __shared__ alignas(16) float s_li[SSZ];

  const int tid = threadIdx.x;
  const int e   = blockIdx.x & (ESZ - 1);       // 0..1023
  const int bg  = blockIdx.x >> 10;             // 0..3
  const int b   = (bg << 2) + (tid >> 6);       // 0..15
  const int s0  = (tid & 63) << 2;              // 0..252 step 4

  // Stage lambda row e (re+im, 2KB) into LDS via the CDNA5 async path;
  // reused by all 4 batch sub-groups in this block.
  async_copy_b32(lr_ + (size_t)e * SSZ + tid, &s_lr[tid]);
  async_copy_b32(li_ + (size_t)e * SSZ + tid, &s_li[tid]);
  wait_async0();
  __syncthreads();

  // Per-(b,e) scalars (L2-hot), hoist delta*x.
  const int be = b * ESZ + e;
  const float dre = dr_[be], dim_ = di_[be];
  const float xre = xr_[be], xim = xi_[be];
  const float dxr = dre * xre - dim_ * xim;
  const float dxi = dre * xim + dim_ * xre;

  // Vector operands.
  const v4f lr4 = *reinterpret_cast<const v4f*>(&s_lr[s0]);   // ds_load_b128
  const v4f li4 = *reinterpret_cast<const v4f*>(&s_li[s0]);
  const v4f br4 = *reinterpret_cast<const v4f*>(Br_ + (size_t)b * SSZ + s0);
  const v4f bi4 = *reinterpret_cast<const v4f*>(Bi_ + (size_t)b * SSZ + s0);

  const size_t hoff = (size_t)be * SSZ + s0;
  // h_prev is a touch-once 33.5MB stream: non-temporal loads (TH=NT).
  const v4f hr4 = __builtin_nontemporal_load(reinterpret_cast<const v4f*>(hr_ + hoff));
  const v4f hi4 = __builtin_nontemporal_load(reinterpret_cast<const v4f*>(hi_ + hoff));

  v4f or4, oi4;
#pragma unroll
  for (int j = 0; j < 4; ++j) {
    // w = delta * lambda  (complex)
    const float wr = dre * lr4[j] - dim_ * li4[j];
    const float wi = dre * li4[j] + dim_ * lr4[j];
    // A' = exp(-w) = exp(-wr) * (cos(wi) - i sin(wi))
    const float ex = __expf(-wr);
    float sn, cs;
    __sincosf(-wi, &sn, &cs);                    // v_sin_f32 / v_cos_f32
    const float Ar = ex * cs;
    const float Ai = ex * sn;
    // additive term: (delta*x) * Bt   (complex)
    const float tr = dxr * br4[j] - dxi * bi4[j];
    const float ti = dxr * bi4[j] + dxi * br4[j];
    // h = A' * h_prev + t
    or4[j] = Ar * hr4[j] - Ai * hi4[j] + tr;
    oi4[j] = Ar * hi4[j] + Ai * hr4[j] + ti;
  }

  // Output is a touch-once stream too: non-temporal stores.
  __builtin_nontemporal_store(or4, reinterpret_cast<v4f*>(out_re + hoff));
  __builtin_nontemporal_store(oi4, reinterpret_cast<v4f*>(out_im + hoff));
}

extern "C" void kernel_launch(void* const* d_in, const int* in_sizes, int n_in,
                              void* d_out, int out_size, void* d_ws, size_t ws_size,
                              hipStream_t stream) {
  // setup_inputs() order:
  const float* xr = (const float*)d_in[0];  // x_gated_re [B,E]
  const float* xi = (const float*)d_in[1];  // x_gated_im
  const float* dr = (const float*)d_in[2];  // delta_re   [B,E]
  const float* di = (const float*)d_in[3];  // delta_im
  const float* Br = (const float*)d_in[4];  // B_re       [B,S]
  const float* Bi = (const float*)d_in[5];  // B_im
  const float* hr = (const float*)d_in[6];  // h_prev_re  [B,E,S]
  const float* hi = (const float*)d_in[7];  // h_prev_im
  const float* lr = (const float*)d_in[8];  // lam_re     [E,S]
  const float* li = (const float*)d_in[9];  // lam_im

  float* out_re = (float*)d_out;                             // [B,E,S]
  float* out_im = out_re + (size_t)BSZ * ESZ * SSZ;          // [B,E,S]

  const dim3 grid((BSZ / 4) * ESZ);   // 4096 blocks: (e, b-group)
  const dim3 block(256);              // 8 wave32
  ssm_update_kernel<<<grid, block, 0, stream>>>(xr, xi, dr, di, Br, Bi,
                                                hr, hi, lr, li, out_re, out_im);
}